// HiPPOLSI_28527172780057
// MI455X (gfx1250) — compile-verified
//
#include <hip/hip_runtime.h>
#include <stdint.h>

// HiPPO-LegS scan, CDNA5 (gfx1250, wave32).
// Phase 1: 128 chunk workgroups compose 128 affine steps each:
//          closed-form O(N)-per-column discretization + f32 WMMA GEMMs in LDS.
// Phase 2: 1 workgroup combines the 128 chunk maps sequentially (L2-resident),
//          staging them via global_load_async_to_lds_b128.

#define N       128
#define BATCH   32
#define CHUNK   128
#define THREADS 256   // 8 wave32 waves

typedef __attribute__((ext_vector_type(2))) float v2f;
typedef __attribute__((ext_vector_type(8))) float v8f;

// Per-wave lower-triangular tile assignment (36 tiles round-robin over 8 waves).
__device__ const int TRI_IB[8][5] = {
  {0,3,5,6,7},{1,3,5,6,7},{1,4,5,6,7},{2,4,5,6,7},
  {2,4,5,7,0},{2,4,6,7,0},{3,4,6,7,0},{3,4,6,7,0}};
__device__ const int TRI_JB[8][5] = {
  {0,2,1,3,4},{0,3,2,4,5},{1,0,3,5,6},{0,1,4,6,7},
  {1,2,5,0,0},{2,3,0,1,0},{0,4,1,2,0},{1,5,2,3,0}};

// A-matrix fragment (16x4 f32) per ISA 7.12.2.
__device__ __forceinline__ v2f frag_a(const float* M, int ld, int r0, int k0, int lane) {
  int r = r0 + (lane & 15);
  int k = k0 + ((lane >> 4) << 1);
  v2f a;
  a.x = M[r * ld + k];
  a.y = M[r * ld + k + 1];
  return a;
}

// B-matrix fragment (4x16 f32): element (k,n); rows striped across lanes.
__device__ __forceinline__ v2f frag_b(const float* M, int ld, int k0, int c0, int lane) {
  int n = c0 + (lane & 15);
  int k = k0 + ((lane >> 4) << 1);
  v2f b;
  b.x = M[k * ld + n];
  b.y = M[(k + 1) * ld + n];
  return b;
}

// C/D 16x16 f32 tile: vgpr v, lanes 0-15 -> row r0+v, lanes 16-31 -> row r0+v+8.
__device__ __forceinline__ void store_tile(float* M, int ld, int r0, int c0, int lane, v8f c) {
  int col = c0 + (lane & 15);
  int rb  = r0 + ((lane >> 4) << 3);
#pragma unroll
  for (int v = 0; v < 8; ++v) M[(rb + v) * ld + col] = c[v];
}

__device__ __forceinline__ v8f wmma4(v2f a, v2f b, v8f c) {
  return __builtin_amdgcn_wmma_f32_16x16x4_f32(false, a, false, b, (short)0, c, false, false);
}

// ---- CDNA5 async LDS<->global copies (ASYNCcnt-tracked, no VGPR round-trip) --
__device__ __forceinline__ void async_load_b128(uint32_t lds_off, const float* gptr) {
  asm volatile("global_load_async_to_lds_b128 %0, %1, off"
               :: "v"(lds_off), "v"(gptr) : "memory");
}
__device__ __forceinline__ void async_store_b128(float* gptr, uint32_t lds_off) {
  asm volatile("global_store_async_from_lds_b128 %0, %1, off"
               :: "v"(gptr), "v"(lds_off) : "memory");
}
__device__ __forceinline__ void wait_async0() {
  asm volatile("s_wait_asynccnt 0x0" ::: "memory");
}

// ---------------------------------------------------------------------------
// Phase 1: per-chunk affine composition.
//   P <- GA_t * P            (128x128 lower-triangular, tile-triangular k)
//   D <- D * GA_t^T + f_t*GB (32x128 state scan from zero)
// GA_t built in LDS each step via rank-1 closed-form forward substitution:
//   A[i,k] = -s_i*s_k (i>k), A[i,i] = -(i+1), s_i = sqrt(2i+1), B[i] = s_i
//   GA = 2*p1^{-1} - I  (alpha = 1/2)
// ---------------------------------------------------------------------------
__global__ __launch_bounds__(THREADS) void hippo_chunk_kernel(
    const float* __restrict__ f, float* __restrict__ Pout,
    float* __restrict__ Dout, int T) {
  __shared__ __align__(16) float GA[N * N];       // 64 KB
  __shared__ __align__(16) float P[N * N];        // 64 KB
  __shared__ __align__(16) float D[BATCH * N];    // 16 KB
  __shared__ float GB[N];
  __shared__ float sl[N];
  __shared__ float fl[BATCH];

  const int tid  = threadIdx.x;
  const int lane = tid & 31;
  const int wave = __builtin_amdgcn_readfirstlane(tid >> 5);  // uniform SGPR
  const int c    = blockIdx.x;
  const int t0   = c * CHUNK;

  if (tid < N) sl[tid] = sqrtf(2.0f * (float)tid + 1.0f);
  for (int i = tid; i < N * N; i += THREADS) {
    int r = i >> 7, cc = i & (N - 1);
    P[i]  = (r == cc) ? 1.0f : 0.0f;
    GA[i] = 0.0f;                       // upper triangle stays zero forever
  }
  for (int i = tid; i < BATCH * N; i += THREADS) D[i] = 0.0f;
  __syncthreads();

  const int npv   = (wave < 4) ? 5 : 4;  // #P-tiles for this wave (scalar)
  const int didx0 = wave * 2;            // 16 D-tiles: 2 batch-blocks x 8 cols

  for (int s = 0; s < CHUNK; ++s) {
    const int   t  = t0 + s;
    const float dt = 1.0f / (float)(t + 1);
    const float h  = 0.5f * dt;

    // ---- Phase A: discretize in O(N) per column (rank-1 structure) ----
    if (tid < N) {
      const int j = tid;              // this thread owns column j of p1^{-1}
      float acc = 0.0f;
      for (int i = j; i < N; ++i) {
        float diag = 1.0f + h * (float)(i + 1);
        float z    = (i == j) ? (1.0f / diag) : (-h * sl[i] * acc / diag);
        acc += sl[i] * z;
        GA[i * N + j] = 2.0f * z - ((i == j) ? 1.0f : 0.0f);
      }
    } else if (tid == N) {
      float acc = 0.0f;               // GB = p1^{-1} (dt*B)
      for (int i = 0; i < N; ++i) {
        float g = sl[i] * (dt - h * acc) / (1.0f + h * (float)(i + 1));
        acc += sl[i] * g;
        GB[i] = g;
      }
    } else if (tid >= 160 && tid < 160 + BATCH) {
      fl[tid - 160] = f[(tid - 160) * T + t];   // f is (BATCH, T, 1)
    }
    __syncthreads();

    // ---- Phase B: all new tiles into registers (reads GA, P, D) ----
    v8f pacc[5];
#pragma unroll
    for (int q = 0; q < 5; ++q) {
      if (q < npv) {                         // uniform (SGPR) guard
        const int ib = TRI_IB[wave][q], jb = TRI_JB[wave][q];
        v8f acc = {};
        for (int kb = jb; kb <= ib; ++kb) {  // uniform triangular k-range
#pragma unroll
          for (int kk = 0; kk < 16; kk += 4) {
            v2f a = frag_a(GA, N, ib * 16, kb * 16 + kk, lane);
            v2f b = frag_b(P,  N, kb * 16 + kk, jb * 16, lane);
            acc = wmma4(a, b, acc);
          }
        }
        pacc[q] = acc;
      }
    }
    v8f dacc[2];
#pragma unroll
    for (int q = 0; q < 2; ++q) {
      const int idx = didx0 + q;
      const int bb = idx >> 3, mb = idx & 7;  // scalar
      v8f acc = {};
      for (int kb = 0; kb <= mb; ++kb) {      // GA^T col-tile mb: k <= m
#pragma unroll
        for (int kk = 0; kk < 16; kk += 4) {
          v2f a = frag_a(D,  N, bb * 16, kb * 16 + kk, lane);
          v2f b = frag_a(GA, N, mb * 16, kb * 16 + kk, lane);  // B-frag of GA^T
          acc = wmma4(a, b, acc);
        }
      }
      const float gb = GB[mb * 16 + (lane & 15)];   // rank-1 input injection
      const int   rb = bb * 16 + ((lane >> 4) << 3);
#pragma unroll
      for (int v = 0; v < 8; ++v) acc[v] += fl[rb + v] * gb;
      dacc[q] = acc;
    }
    __syncthreads();

    // ---- Phase C: in-place write-back ----
#pragma unroll
    for (int q = 0; q < 5; ++q)
      if (q < npv)
        store_tile(P, N, TRI_IB[wave][q] * 16, TRI_JB[wave][q] * 16, lane, pacc[q]);
#pragma unroll
    for (int q = 0; q < 2; ++q) {
      const int idx = didx0 + q;
      store_tile(D, N, (idx >> 3) * 16, (idx & 7) * 16, lane, dacc[q]);
    }
    __syncthreads();
  }

  // Drain chunk results LDS -> global via async stores (STORE path, b128/lane).
  {
    const uint32_t p_base = (uint32_t)(uintptr_t)&P[0];
    const uint32_t d_base = (uint32_t)(uintptr_t)&D[0];
    for (int i = tid * 4; i < N * N; i += THREADS * 4)
      async_store_b128(Pout + (size_t)c * (N * N) + i, p_base + (uint32_t)i * 4u);
    for (int i = tid * 4; i < BATCH * N; i += THREADS * 4)
      async_store_b128(Dout + (size_t)c * (BATCH * N) + i, d_base + (uint32_t)i * 4u);
    wait_async0();
  }
}

// ---------------------------------------------------------------------------
// Phase 2: sequential combine over chunk maps:  C <- C * P_c^T + D_c
// P_c streams from L2 (10.5 MB total << 192 MB L2); async-load into LDS and
// prefetch the next chunk's P while computing.
// ---------------------------------------------------------------------------
__global__ __launch_bounds__(THREADS) void hippo_combine_kernel(
    const float* __restrict__ Pout, const float* __restrict__ Dout,
    float* __restrict__ out, int nchunks) {
  __shared__ __align__(16) float Pl[N * N];       // 64 KB
  __shared__ __align__(16) float Cl[BATCH * N];   // 16 KB
  __shared__ __align__(16) float Dl[BATCH * N];   // 16 KB

  const int tid  = threadIdx.x;
  const int lane = tid & 31;
  const int wave = __builtin_amdgcn_readfirstlane(tid >> 5);  // uniform SGPR

  const uint32_t pl_base = (uint32_t)(uintptr_t)&Pl[0];
  const uint32_t dl_base = (uint32_t)(uintptr_t)&Dl[0];

  for (int i = tid; i < BATCH * N; i += THREADS) Cl[i] = 0.0f;
  __syncthreads();

  for (int c = 0; c < nchunks; ++c) {
    // Async global->LDS staging of this chunk's (P_c, D_c).
    for (int i = tid * 4; i < N * N; i += THREADS * 4)
      async_load_b128(pl_base + (uint32_t)i * 4u, Pout + (size_t)c * (N * N) + i);
    for (int i = tid * 4; i < BATCH * N; i += THREADS * 4)
      async_load_b128(dl_base + (uint32_t)i * 4u, Dout + (size_t)c * (BATCH * N) + i);
    if (c + 1 < nchunks)   // one 256B line per thread -> covers next P_c
      __builtin_prefetch(&Pout[(size_t)(c + 1) * (N * N) + tid * 64], 0, 1);
    wait_async0();
    __syncthreads();

    v8f dacc[2];
#pragma unroll
    for (int q = 0; q < 2; ++q) {
      const int idx = wave * 2 + q;
      const int bb = idx >> 3, mb = idx & 7;  // scalar
      v8f acc = {};
      for (int kb = 0; kb <= mb; ++kb) {      // P_c lower-tri: k <= m
#pragma unroll
        for (int kk = 0; kk < 16; kk += 4) {
          v2f a = frag_a(Cl, N, bb * 16, kb * 16 + kk, lane);
          v2f b = frag_a(Pl, N, mb * 16, kb * 16 + kk, lane);  // B-frag of P^T
          acc = wmma4(a, b, acc);
        }
      }
      const int col = mb * 16 + (lane & 15);
      const int rb  = bb * 16 + ((lane >> 4) << 3);
#pragma unroll
      for (int v = 0; v < 8; ++v) acc[v] += Dl[(rb + v) * N + col];
      dacc[q] = acc;
    }
    __syncthreads();
#pragma unroll
    for (int q = 0; q < 2; ++q) {
      const int idx = wave * 2 + q;
      store_tile(Cl, N, (idx >> 3) * 16, (idx & 7) * 16, lane, dacc[q]);
    }
    __syncthreads();
  }

  for (int i = tid; i < BATCH * N; i += THREADS) out[i] = Cl[i];  // (batch,1,N)
}

extern "C" void kernel_launch(void* const* d_in, const int* in_sizes, int n_in,
                              void* d_out, int out_size, void* d_ws, size_t ws_size,
                              hipStream_t stream) {
  (void)n_in; (void)out_size; (void)ws_size;
  const float* f = (const float*)d_in[0];   // (BATCH, T, 1) fp32
  const int T       = in_sizes[0] / BATCH;  // 16384
  const int nchunks = T / CHUNK;            // 128

  // Workspace: P_c (nchunks*128*128 f32 = 8 MB) + D_c (nchunks*32*128 f32 = 2 MB)
  float* Pout = (float*)d_ws;
  float* Dout = Pout + (size_t)nchunks * N * N;

  hippo_chunk_kernel<<<nchunks, THREADS, 0, stream>>>(f, Pout, Dout, T);
  hippo_combine_kernel<<<1, THREADS, 0, stream>>>(Pout, Dout, (float*)d_out, nchunks);
}